// SpikeTraceComputer_87557203296888
// MI455X (gfx1250) — compile-verified
//
#include <hip/hip_runtime.h>

// Spike trace EMA: r_t = a*r_{t-1} + (1-a)*s_t over T, for (B,H) lanes.
// Chunked linear-recurrence formulation on f32 WMMA (16x16x4), wave32.
//
// Per wave: one 16-channel group, one T-segment. Each 16-step chunk:
//   D(16x16) = sum_{j=0..3} A_j(16x4) x B_j(4x16)  +  A_c x B_c(carry)
// A_j[t][k] = (t>=s) ? (1-a)*a^(t-s) : 0   with s = 4j+k   (constants)
// A_c[t][0] = a^(t+1)                                       (carry column)
// carry' = D[15][:]  (VGPR7, lanes 16..31 -> shuffled to lanes 0..15)

typedef float v2f __attribute__((ext_vector_type(2)));
typedef float v8f __attribute__((ext_vector_type(8)));

#define ALPHA 0.9f

constexpr int Bb = 16;
constexpr int Tt = 2048;
constexpr int Hh = 2048;
constexpr int CH = 16;                     // timesteps per WMMA chunk
constexpr int SEG_CHUNKS = 32;             // 512 timesteps per segment
constexpr int WARM_CHUNKS = 12;            // 192-step warmup: a^192 ~ 1.7e-9 < f32 eps
constexpr int NSEG = Tt / (CH * SEG_CHUNKS);        // 4
constexpr int HGROUPS = Hh / 16;                    // 128
constexpr int NWAVES = Bb * HGROUPS * NSEG;         // 8192 waves
constexpr int BLOCK = 256;                          // 8 waves per block

__global__ __launch_bounds__(BLOCK, 1)
void spike_trace_wmma(const float* __restrict__ S, float* __restrict__ O) {
  const int wave = (int)((blockIdx.x * blockDim.x + threadIdx.x) >> 5);
  const int lane = (int)(threadIdx.x & 31);
  const int row  = lane & 15;       // M (A/D rows) or N (B/D cols) index
  const bool hi  = lane >= 16;

  // wave -> (seg, channel-group, batch)
  const int seg = wave & (NSEG - 1);
  const int hg  = (wave >> 2) & (HGROUPS - 1);
  const int b   = wave >> 9;
  const int h0  = hg * 16;

  // ---- constant A tiles (per-lane, computed once) ----
  // A 16x4 f32 layout: lanes 0-15 hold K={0,1} in {v0,v1}; lanes 16-31 hold K={2,3}.
  v2f a[4];
#pragma unroll
  for (int j = 0; j < 4; ++j) {
    const int k = 4 * j + (hi ? 2 : 0);
    a[j].x = (row >= k)     ? (1.0f - ALPHA) * powf(ALPHA, (float)(row - k))     : 0.0f;
    a[j].y = (row >= k + 1) ? (1.0f - ALPHA) * powf(ALPHA, (float)(row - k - 1)) : 0.0f;
  }
  // Carry column: A_c[t][0] = a^(t+1); columns 1..3 zero.
  v2f ac;
  ac.x = hi ? 0.0f : powf(ALPHA, (float)(row + 1));
  ac.y = 0.0f;

  const size_t base = (size_t)b * Tt * Hh + (size_t)h0 + (size_t)row;
  const float* Sb = S + base;   // + t*Hh addresses spikes[b][t][h0+row]
  float*       Ob = O + base;

  const int t_seg = seg * (CH * SEG_CHUNKS);
  float carry = 0.0f;

  for (int c = -WARM_CHUNKS; c < SEG_CHUNKS; ++c) {
    const int t0 = t_seg + c * CH;
    if (t0 < 0) continue;       // segment 0 starts exactly (carry = 0)

    // carry term: B_c row K=0 holds carry[N], other rows zero
    v2f bc;
    bc.x = hi ? 0.0f : carry;
    bc.y = 0.0f;
    v8f d = {};
    d = __builtin_amdgcn_wmma_f32_16x16x4_f32(false, ac, false, bc,
                                              (short)0, d, false, false);

    // 4 chained K=4 sub-chunks of the 16x16 lower-triangular multiply
#pragma unroll
    for (int j = 0; j < 4; ++j) {
      const int kb = 4 * j + (hi ? 2 : 0);
      const float* p = Sb + (size_t)(t0 + kb) * Hh;
      v2f bj;
      bj.x = p[0];              // row K = kb
      bj.y = p[Hh];             // row K = kb+1
      if (c < SEG_CHUNKS - 1)   // in-bounds only (WGP prefetch is non-speculative)
        __builtin_prefetch(p + (size_t)CH * Hh, 0, 1);
      d = __builtin_amdgcn_wmma_f32_16x16x4_f32(false, a[j], false, bj,
                                                (short)0, d, false, false);
    }

    // carry' = D[M=15][N] : lives in d.s7 of lanes 16..31; move to lanes 0..15
    carry = __shfl(d.s7, 16 + row, 32);

    if (c >= 0) {
      // D layout: VGPR i holds M = i + (hi?8:0), N = row
      float* q = Ob + (size_t)(t0 + (hi ? 8 : 0)) * Hh;
#pragma unroll
      for (int i = 0; i < 8; ++i)
        q[(size_t)i * Hh] = d[i];
    }
  }
}

extern "C" void kernel_launch(void* const* d_in, const int* in_sizes, int n_in,
                              void* d_out, int out_size, void* d_ws, size_t ws_size,
                              hipStream_t stream) {
  (void)in_sizes; (void)n_in; (void)d_ws; (void)ws_size; (void)out_size;
  const float* spikes = (const float*)d_in[0];
  float* out = (float*)d_out;

  const int threads = NWAVES * 32;             // 262144
  dim3 grid(threads / BLOCK), block(BLOCK);
  spike_trace_wmma<<<grid, block, 0, stream>>>(spikes, out);
}